// CG_loss_56375740727695
// MI455X (gfx1250) — compile-verified
//
#include <hip/hip_runtime.h>
#include <math.h>

#define N128   128
#define NENT   8256
#define KAPPA  0.276f
#define MAXITER 20

typedef float v2f __attribute__((ext_vector_type(2)));
typedef float v8f __attribute__((ext_vector_type(8)));

// ---------------- complex helpers (float2 = re,im) ----------------
__device__ __forceinline__ float2 cmul(float2 a, float2 b) {
    return make_float2(a.x * b.x - a.y * b.y, a.x * b.y + a.y * b.x);
}
__device__ __forceinline__ float2 cjmul(float2 u, float2 b) { // conj(u)*b
    return make_float2(u.x * b.x + u.y * b.y, u.x * b.y - u.y * b.x);
}
__device__ __forceinline__ float2 cadd(float2 a, float2 b) {
    return make_float2(a.x + b.x, a.y + b.y);
}
__device__ __forceinline__ float2 csub(float2 a, float2 b) {
    return make_float2(a.x - b.x, a.y - b.y);
}
__device__ __forceinline__ float2 cdiv(float2 a, float2 b) {
    float inv = 1.0f / (b.x * b.x + b.y * b.y);
    return make_float2((a.x * b.x + a.y * b.y) * inv, (a.y * b.x - a.x * b.y) * inv);
}

// Packed lower-triangular load with UNCONDITIONAL, in-bounds address math:
// column clamped with min() so the address is always valid (v_min_u32),
// only the loaded value is selected against zero (v_cndmask). No EXEC changes.
__device__ __forceinline__ float triload(const float* __restrict__ Lp, int row, int col) {
    const int c   = (col < row) ? col : row;              // v_min
    const int idx = (((row * (row + 1)) >> 1) + c);       // always <= tri(row)+row < NENT
    const float v = Lp[idx];
    return (col <= row) ? v : 0.0f;                       // v_cndmask
}

// wave32 butterfly sum
__device__ __forceinline__ float2 wave_sum(float2 v) {
    #pragma unroll
    for (int off = 16; off > 0; off >>= 1) {
        v.x += __shfl_xor(v.x, off, 32);
        v.y += __shfl_xor(v.y, off, 32);
    }
    return v;
}

// Wilson stencil: out = in - kappa * sum_mu [(I + s*G_mu) fwd + (I - s*G_mu) bwd]
// s = -1 -> D, s = +1 -> D^dagger.   Sites handled by threads t<64 (2 spins each).
__device__ void dirac(const float2* __restrict__ in, float2* __restrict__ out,
                      const float2* __restrict__ U, int t, float s) {
    if (t < 64) {
        int i = t >> 3, j = t & 7;
        int ip = (i + 1) & 7, im = (i - 1) & 7;
        int jp = (j + 1) & 7, jm = (j - 1) & 7;
        float2 x0 = in[2 * t], x1 = in[2 * t + 1];

        // mu = 0 (axis i)
        float2 u0  = U[(i << 3) | j];
        float2 u0b = U[(im << 3) | j];
        float2 f0 = cmul(u0, in[2 * ((ip << 3) | j)]);
        float2 f1 = cmul(u0, in[2 * ((ip << 3) | j) + 1]);
        float2 b0 = cjmul(u0b, in[2 * ((im << 3) | j)]);
        float2 b1 = cjmul(u0b, in[2 * ((im << 3) | j) + 1]);
        float2 a0 = make_float2(f0.x + s * f1.x + b0.x - s * b1.x,
                                f0.y + s * f1.y + b0.y - s * b1.y);
        float2 a1 = make_float2(f1.x + s * f0.x + b1.x - s * b0.x,
                                f1.y + s * f0.y + b1.y - s * b0.y);

        // mu = 1 (axis j)
        float2 u1  = U[64 + ((i << 3) | j)];
        float2 u1b = U[64 + ((i << 3) | jm)];
        f0 = cmul(u1, in[2 * ((i << 3) | jp)]);
        f1 = cmul(u1, in[2 * ((i << 3) | jp) + 1]);
        b0 = cjmul(u1b, in[2 * ((i << 3) | jm)]);
        b1 = cjmul(u1b, in[2 * ((i << 3) | jm) + 1]);
        a0.x += f0.x + s * f1.y;  a0.y += f0.y - s * f1.x;
        a1.x += f1.x - s * f0.y;  a1.y += f1.y + s * f0.x;
        a0.x += b0.x - s * b1.y;  a0.y += b0.y + s * b1.x;
        a1.x += b1.x + s * b0.y;  a1.y += b1.y - s * b0.x;

        out[2 * t]     = make_float2(x0.x - KAPPA * a0.x, x0.y - KAPPA * a0.y);
        out[2 * t + 1] = make_float2(x1.x - KAPPA * a1.x, x1.y - KAPPA * a1.y);
    }
}

// z = M * v (complex 128x128). 256 threads: row = t&127, half = t>>7 sums 64 cols.
// float4 (ds_load_b128) panels with a rotated block index: start banks of the 16
// rotations cover all 64 LDS banks across each 16-lane group.
__device__ void matvec(const float* __restrict__ Mr, const float* __restrict__ Mi,
                       const float2* __restrict__ v, float2* __restrict__ out,
                       float2* __restrict__ red, int t) {
    const int row = t & 127, half = t >> 7;
    const float4* __restrict__ Mr4 = (const float4*)(Mr + (row << 7) + (half << 6));
    const float4* __restrict__ Mi4 = (const float4*)(Mi + (row << 7) + (half << 6));
    const float4* __restrict__ vb4 = (const float4*)(v + (half << 6));
    float2 acc = make_float2(0.f, 0.f);
    #pragma unroll
    for (int cc = 0; cc < 16; ++cc) {
        const int blk = (cc + t) & 15;
        float4 mr = Mr4[blk];
        float4 mi = Mi4[blk];
        float4 va = vb4[2 * blk];       // (v0.re v0.im v1.re v1.im)
        float4 vb = vb4[2 * blk + 1];   // (v2.re v2.im v3.re v3.im)
        acc.x += mr.x * va.x - mi.x * va.y;  acc.y += mr.x * va.y + mi.x * va.x;
        acc.x += mr.y * va.z - mi.y * va.w;  acc.y += mr.y * va.w + mi.y * va.z;
        acc.x += mr.z * vb.x - mi.z * vb.y;  acc.y += mr.z * vb.y + mi.z * vb.x;
        acc.x += mr.w * vb.z - mi.w * vb.w;  acc.y += mr.w * vb.w + mi.w * vb.z;
    }
    red[t] = acc;
    __syncthreads();
    if (t < 128) out[t] = cadd(red[t], red[t + 128]);
    __syncthreads();
}

// complex dot <a,c> = sum conj(a)*c, broadcast to all threads (2 barriers)
__device__ float2 cdot(const float2* __restrict__ a, const float2* __restrict__ c,
                       float2* __restrict__ red, int t) {
    float2 term = make_float2(0.f, 0.f);
    if (t < 128) {
        float2 av = a[t], cv = c[t];
        term = make_float2(av.x * cv.x + av.y * cv.y, av.x * cv.y - av.y * cv.x);
    }
    term = wave_sum(term);
    if ((t & 31) == 0) red[t >> 5] = term;
    __syncthreads();
    float2 res = make_float2(0.f, 0.f);
    #pragma unroll
    for (int w = 0; w < 8; ++w) res = cadd(res, red[w]);
    __syncthreads();
    return res;
}

__global__ void zero_out_kernel(float* out) {
    if (threadIdx.x == 0 && blockIdx.x == 0) out[0] = 0.0f;
}

// One workgroup (256 threads = 8 waves) per batch element.
__global__ void __launch_bounds__(256)
cg_wilson_kernel(const float* __restrict__ nre, const float* __restrict__ nim,
                 const float* __restrict__ theta, const float* __restrict__ bglob,
                 float* __restrict__ out, float invB) {
    extern __shared__ unsigned char smem[];
    float*  Mr   = (float*)smem;            // 128x128 = 64 KB
    float*  Mi   = Mr + 16384;              // 64 KB
    float2* Uld  = (float2*)(Mi + 16384);   // 128 links
    float2* bvec = Uld + 128;
    float2* xv   = bvec + 128;
    float2* rv   = xv + 128;
    float2* pv   = rv + 128;
    float2* zv   = pv + 128;
    float2* apv  = zv + 128;
    float2* tv   = apv + 128;
    float2* red  = tv + 128;                // 256 entries

    const int bid = blockIdx.x;
    const int t   = threadIdx.x;
    const float* __restrict__ Lre = nre + (size_t)bid * NENT;
    const float* __restrict__ Lim = nim + (size_t)bid * NENT;

    // ---------------- Phase 1: M = L^H L via f32 WMMA (batched Gram GEMM) ----
    // Wave i owns 16-row tile band i; loops output tiles j = 0..7.
    // A[m,kk] = L[k+kk, 16i+m], B[kk,n] = L[k+kk, 16j+n] per ISA 32-bit layouts.
    {
        const int lane = t & 31;
        const int i    = t >> 5;              // wave id = tile row
        const int m    = lane & 15;
        const int ks   = (lane >> 4) << 1;    // K pair select: 0 or 2
        for (int j = 0; j < 8; ++j) {
            v8f cr = {0.f, 0.f, 0.f, 0.f, 0.f, 0.f, 0.f, 0.f};
            v8f c1 = cr, c2 = cr;
            const int ca = (i << 4) + m;
            const int cb = (j << 4) + m;
            const int k0 = ((i > j) ? i : j) << 4;   // rows < 16*max(i,j) are all zero
            const int k1 = k0 + 16;                  // guards only needed in [k0, k1)

            // --- prologue: 4 guarded (branchless) k-steps across the diagonal block
            #pragma unroll
            for (int k = k0; k < k1; k += 4) {
                const int r0 = k + ks, r1 = r0 + 1;
                v2f ar, ai, br, bi;
                ar.x = triload(Lre, r0, ca); ar.y = triload(Lre, r1, ca);
                ai.x = triload(Lim, r0, ca); ai.y = triload(Lim, r1, ca);
                br.x = triload(Lre, r0, cb); br.y = triload(Lre, r1, cb);
                bi.x = triload(Lim, r0, cb); bi.y = triload(Lim, r1, cb);
                cr = __builtin_amdgcn_wmma_f32_16x16x4_f32(false, ar, false, br, (short)0, cr, false, false);
                cr = __builtin_amdgcn_wmma_f32_16x16x4_f32(false, ai, false, bi, (short)0, cr, false, false);
                c1 = __builtin_amdgcn_wmma_f32_16x16x4_f32(false, ar, false, bi, (short)0, c1, false, false);
                c2 = __builtin_amdgcn_wmma_f32_16x16x4_f32(false, ai, false, br, (short)0, c2, false, false);
            }

            // --- steady state: fully in-range, pure loads + WMMA, incremental
            //     triangular offsets (tri(r+4) = tri(r) + 4r + 10, no multiplies);
            //     unroll 2 so the scheduler can pipeline loads over WMMAs.
            int r0   = k1 + ks;
            int off0 = (r0 * (r0 + 1)) >> 1;
            #pragma unroll 2
            for (int k = k1; k < 128; k += 4) {
                const int off1 = off0 + r0 + 1;
                v2f ar, ai, br, bi;
                ar.x = Lre[off0 + ca]; ar.y = Lre[off1 + ca];
                ai.x = Lim[off0 + ca]; ai.y = Lim[off1 + ca];
                br.x = Lre[off0 + cb]; br.y = Lre[off1 + cb];
                bi.x = Lim[off0 + cb]; bi.y = Lim[off1 + cb];
                // Re(M) += Lr^T Lr + Li^T Li ; Im(M) = Lr^T Li - Li^T Lr (c1 - c2)
                cr = __builtin_amdgcn_wmma_f32_16x16x4_f32(false, ar, false, br, (short)0, cr, false, false);
                cr = __builtin_amdgcn_wmma_f32_16x16x4_f32(false, ai, false, bi, (short)0, cr, false, false);
                c1 = __builtin_amdgcn_wmma_f32_16x16x4_f32(false, ar, false, bi, (short)0, c1, false, false);
                c2 = __builtin_amdgcn_wmma_f32_16x16x4_f32(false, ai, false, br, (short)0, c2, false, false);
                off0 += (r0 << 2) + 10;
                r0   += 4;
            }

            // C/D layout: VGPR v -> row v (lanes 0-15) or v+8 (lanes 16-31)
            const int rbase = (i << 4) + ((lane >> 4) << 3);
            const int col   = (j << 4) + m;
            #pragma unroll
            for (int v = 0; v < 8; ++v) {
                Mr[((rbase + v) << 7) + col] = cr[v];
                Mi[((rbase + v) << 7) + col] = c1[v] - c2[v];
            }
        }
    }

    // ---------------- Phase 2 init: U = exp(i*theta), r0 = b, x0 = 0 --------
    if (t < 128) {
        float th = theta[(size_t)bid * 128 + t];
        float sn, cs;
        sincosf(th, &sn, &cs);
        Uld[t]  = make_float2(cs, sn);
        bvec[t] = make_float2(bglob[(size_t)bid * 128 + t], 0.f);
        xv[t]   = make_float2(0.f, 0.f);
        rv[t]   = bvec[t];
    }
    __syncthreads();

    matvec(Mr, Mi, rv, zv, red, t);           // z = M r
    if (t < 128) pv[t] = zv[t];
    __syncthreads();
    float2 rz = cdot(rv, zv, red, t);

    // ---------------- 20 CG iterations entirely in LDS ----------------------
    for (int it = 0; it < MAXITER; ++it) {
        dirac(pv, tv, Uld, t, -1.f);  __syncthreads();   // t = D p
        dirac(tv, apv, Uld, t, 1.f);  __syncthreads();   // Ap = D^dag t
        float2 pAp   = cdot(pv, apv, red, t);
        float2 alpha = cdiv(rz, pAp);
        if (t < 128) {
            xv[t] = cadd(xv[t], cmul(alpha, pv[t]));
            rv[t] = csub(rv[t], cmul(alpha, apv[t]));
        }
        __syncthreads();
        matvec(Mr, Mi, rv, zv, red, t);                  // z = M r
        float2 rz2  = cdot(rv, zv, red, t);
        float2 beta = cdiv(rz2, rz);
        if (t < 128) pv[t] = cadd(zv[t], cmul(beta, pv[t]));
        rz = rz2;
        __syncthreads();
    }

    // ---------------- residual ||A x - b||, mean over batch ------------------
    dirac(xv, tv, Uld, t, -1.f);  __syncthreads();
    dirac(tv, apv, Uld, t, 1.f);  __syncthreads();
    float nrm = 0.f;
    if (t < 128) {
        float2 d = csub(apv[t], bvec[t]);
        nrm = d.x * d.x + d.y * d.y;
    }
    #pragma unroll
    for (int off = 16; off > 0; off >>= 1) nrm += __shfl_xor(nrm, off, 32);
    if ((t & 31) == 0) red[t >> 5].x = nrm;
    __syncthreads();
    if (t == 0) {
        float s = 0.f;
        #pragma unroll
        for (int w = 0; w < 8; ++w) s += red[w].x;
        atomicAdd(out, sqrtf(s) * invB);
    }
}

extern "C" void kernel_launch(void* const* d_in, const int* in_sizes, int n_in,
                              void* d_out, int out_size, void* d_ws, size_t ws_size,
                              hipStream_t stream) {
    (void)n_in; (void)out_size; (void)d_ws; (void)ws_size;
    const float* nre   = (const float*)d_in[0];
    const float* nim   = (const float*)d_in[1];
    const float* theta = (const float*)d_in[2];
    const float* bglob = (const float*)d_in[3];
    float* out = (float*)d_out;

    const int B = in_sizes[0] / NENT;
    // LDS: Mr + Mi (128 KB) + U/vectors/reduction (10 KB) = 141312 B
    const size_t smem_bytes = 2 * 16384 * sizeof(float) + (128 * 8 + 256) * sizeof(float2);

    zero_out_kernel<<<1, 64, 0, stream>>>(out);
    cg_wilson_kernel<<<B, 256, smem_bytes, stream>>>(nre, nim, theta, bglob, out,
                                                     1.0f / (float)B);
}